// PretrainedModel_85899345920584
// MI455X (gfx1250) — compile-verified
//
#include <hip/hip_runtime.h>
#include <hip/hip_bf16.h>
#include <cstdint>
#include <cstddef>

// ---------------------------------------------------------------------------
// GCN forward for MI455X (gfx1250, wave32).
// GEMMs use native f32 WMMA: v_wmma_f32_16x16x4_f32.
// Edge aggregation uses global_atomic_add_f32 (stays in 192MB L2).
// ---------------------------------------------------------------------------

typedef __attribute__((ext_vector_type(2))) float v2f;
typedef __attribute__((ext_vector_type(8))) float v8f;

#define N_NODES 100000
#define N_EDGES 1600000
#define D_INF   128
#define D_HIDF  128
#define D_OUTF  64

// ---- generic float4 zero fill -------------------------------------------
__global__ void zero_f4_kernel(float4* __restrict__ p, int n4) {
    int i = blockIdx.x * blockDim.x + threadIdx.x;
    if (i < n4) p[i] = make_float4(0.f, 0.f, 0.f, 0.f);
}

// ---- degree (integer atomics, deterministic) ----------------------------
__global__ void deg_count_kernel(const int* __restrict__ dst, int* __restrict__ deg, int e) {
    int i = blockIdx.x * blockDim.x + threadIdx.x;
    if (i < e) atomicAdd(&deg[dst[i]], 1);
}

__global__ void dinv_kernel(const int* __restrict__ deg, float* __restrict__ dinv, int n) {
    int i = blockIdx.x * blockDim.x + threadIdx.x;
    if (i < n) dinv[i] = rsqrtf((float)deg[i] + 1.0f);   // +1 = self loop
}

// ---- WMMA f32 GEMM: OUT[nRows x nCols] = X[nRows x 128] @ W[128 x nCols] -
// Block = 256 threads (8 waves), block tile 64x32, wave tile 16x16.
#define BM 64
#define BN 32
#define BK 128
#define AS_STRIDE 132   // padded: bank-conflict-free column reads, 16B aligned
#define WS_STRIDE 36    // padded: 16B aligned rows

__global__ __launch_bounds__(256)
void gemm_wmma_f32_kernel(const float* __restrict__ X, const float* __restrict__ W,
                          float* __restrict__ OUT, const float* __restrict__ bias,
                          int nRows, int nCols) {
    __shared__ float As[BM * AS_STRIDE];
    __shared__ float Ws[BK * WS_STRIDE];

    const int tid     = threadIdx.x;
    const int rowBase = blockIdx.y * BM;
    const int colBase = blockIdx.x * BN;

    // cooperative load X tile (64x128) : 2048 float4, clamped rows
    for (int i = tid; i < (BM * BK) / 4; i += 256) {
        int r  = i >> 5;            // 32 float4 per row
        int c4 = (i & 31) << 2;
        int gr = rowBase + r;
        if (gr >= nRows) gr = nRows - 1;
        float4 v = *(const float4*)(X + (size_t)gr * BK + c4);
        *(float4*)(&As[r * AS_STRIDE + c4]) = v;
    }
    // cooperative load W tile (128x32) : 1024 float4
    for (int i = tid; i < (BK * BN) / 4; i += 256) {
        int r  = i >> 3;            // 8 float4 per row
        int c4 = (i & 7) << 2;
        float4 v = *(const float4*)(W + (size_t)r * nCols + colBase + c4);
        *(float4*)(&Ws[r * WS_STRIDE + c4]) = v;
    }
    __syncthreads();

    const int lane = tid & 31;
    const int wave = tid >> 5;
    const int wRow = (wave & 3) * 16;    // 4 wave-rows
    const int wCol = (wave >> 2) * 16;   // 2 wave-cols
    const int half = lane >> 4;          // lane half selects K pair / M half
    const int l15  = lane & 15;

    v8f c = {};
    #pragma unroll
    for (int k0 = 0; k0 < BK; k0 += 4) {
        const int ka = k0 + 2 * half;
        // A 16x4 f32 frag: lane holds row M=l15; vgpr0 K=ka, vgpr1 K=ka+1
        v2f a;
        a.x = As[(wRow + l15) * AS_STRIDE + ka];
        a.y = As[(wRow + l15) * AS_STRIDE + ka + 1];
        // B 4x16 f32 frag: lane holds col N=l15; vgpr0 K=ka, vgpr1 K=ka+1
        v2f b;
        b.x = Ws[(ka)     * WS_STRIDE + wCol + l15];
        b.y = Ws[(ka + 1) * WS_STRIDE + wCol + l15];
        c = __builtin_amdgcn_wmma_f32_16x16x4_f32(
                /*neg_a=*/false, a, /*neg_b=*/false, b,
                /*c_mod=*/(short)0, c, /*reuse_a=*/false, /*reuse_b=*/false);
    }

    // C/D layout: vgpr v -> row (half? 8+v : v), col = l15
    const int gc = colBase + wCol + l15;
    #pragma unroll
    for (int v = 0; v < 8; ++v) {
        int gr = rowBase + wRow + half * 8 + v;
        if (gr < nRows) {
            float val = c[v];
            if (bias) val += bias[gc];
            OUT[(size_t)gr * nCols + gc] = val;
        }
    }
}

// ---- edge scatter: one wave per edge, one float4 per lane ---------------
__global__ __launch_bounds__(256)
void scatter_kernel(const float* __restrict__ h, const int* __restrict__ src,
                    const int* __restrict__ dst, const float* __restrict__ dinv,
                    float* __restrict__ out, int e) {
    int g    = blockIdx.x * blockDim.x + threadIdx.x;
    int edge = g >> 5;
    if (edge >= e) return;
    int lane = g & 31;
    int s = src[edge];
    int d = dst[edge];
    float norm = dinv[s] * dinv[d];
    float4 v = *(const float4*)(h + (size_t)s * D_HIDF + lane * 4);
    float* o = out + (size_t)d * D_HIDF + lane * 4;
    atomicAdd(o + 0, v.x * norm);
    atomicAdd(o + 1, v.y * norm);
    atomicAdd(o + 2, v.z * norm);
    atomicAdd(o + 3, v.w * norm);
}

// ---- fused self-loop + bias + relu (float4 granularity) -----------------
__global__ void sl_bias_relu_kernel(float* __restrict__ agg, const float* __restrict__ h,
                                    const float* __restrict__ dinv,
                                    const float* __restrict__ bias, int n) {
    int i4 = blockIdx.x * blockDim.x + threadIdx.x;   // float4 index
    if (i4 >= n * (D_HIDF / 4)) return;
    int v = i4 >> 5;            // node
    int f = (i4 & 31) << 2;     // feature base
    float di = dinv[v];
    float sl = di * di;
    float4 a  = *(float4*)(agg + (size_t)v * D_HIDF + f);
    float4 hh = *(const float4*)(h + (size_t)v * D_HIDF + f);
    float4 b  = *(const float4*)(bias + f);
    a.x = fmaxf(a.x + hh.x * sl + b.x, 0.f);
    a.y = fmaxf(a.y + hh.y * sl + b.y, 0.f);
    a.z = fmaxf(a.z + hh.z * sl + b.z, 0.f);
    a.w = fmaxf(a.w + hh.w * sl + b.w, 0.f);
    *(float4*)(agg + (size_t)v * D_HIDF + f) = a;
}

// ---------------------------------------------------------------------------
extern "C" void kernel_launch(void* const* d_in, const int* in_sizes, int n_in,
                              void* d_out, int out_size, void* d_ws, size_t ws_size,
                              hipStream_t stream) {
    const float* x    = (const float*)d_in[0];
    const int*   eidx = (const int*)d_in[1];    // [2, E]
    const float* W1   = (const float*)d_in[2];
    const float* b1   = (const float*)d_in[3];
    const float* W2   = (const float*)d_in[4];
    const float* b2   = (const float*)d_in[5];
    const float* Wd   = (const float*)d_in[6];
    const float* bd   = (const float*)d_in[7];
    const int* src = eidx;
    const int* dst = eidx + N_EDGES;

    // workspace carve-out (256B aligned)
    char*  ws  = (char*)d_ws;
    size_t off = 0;
    auto carve = [&](size_t bytes) -> void* {
        void* p = ws + off;
        off += (bytes + 255) & ~(size_t)255;
        return p;
    };
    int*   deg  = (int*)  carve((size_t)N_NODES * 4);
    float* dinv = (float*)carve((size_t)N_NODES * 4);
    float* bufA = (float*)carve((size_t)N_NODES * D_HIDF * 4);
    float* bufB = (float*)carve((size_t)N_NODES * D_HIDF * 4);

    const int feat4  = N_NODES * (D_HIDF / 4);            // 3.2M float4
    const int scatTh = N_EDGES * 32;                      // one wave per edge

    // degrees -> dinv
    zero_f4_kernel<<<(N_NODES / 4 + 255) / 256, 256, 0, stream>>>((float4*)deg, N_NODES / 4);
    deg_count_kernel<<<(N_EDGES + 255) / 256, 256, 0, stream>>>(dst, deg, N_EDGES);
    dinv_kernel<<<(N_NODES + 255) / 256, 256, 0, stream>>>(deg, dinv, N_NODES);

    dim3 blk(256);
    dim3 gHid((D_HIDF + BN - 1) / BN, (N_NODES + BM - 1) / BM);
    dim3 gOut((D_OUTF + BN - 1) / BN, (N_NODES + BM - 1) / BM);

    // layer 1: bufA = x @ W1 ; bufB = aggregate(bufA) ; bufB = relu(bufB + sl + b1)
    gemm_wmma_f32_kernel<<<gHid, blk, 0, stream>>>(x, W1, bufA, nullptr, N_NODES, D_HIDF);
    zero_f4_kernel<<<(feat4 + 255) / 256, 256, 0, stream>>>((float4*)bufB, feat4);
    scatter_kernel<<<(scatTh + 255) / 256, 256, 0, stream>>>(bufA, src, dst, dinv, bufB, N_EDGES);
    sl_bias_relu_kernel<<<(feat4 + 255) / 256, 256, 0, stream>>>(bufB, bufA, dinv, b1, N_NODES);

    // layer 2: bufA = bufB @ W2 ; bufB = aggregate(bufA) ; relu(+b2)
    gemm_wmma_f32_kernel<<<gHid, blk, 0, stream>>>(bufB, W2, bufA, nullptr, N_NODES, D_HIDF);
    zero_f4_kernel<<<(feat4 + 255) / 256, 256, 0, stream>>>((float4*)bufB, feat4);
    scatter_kernel<<<(scatTh + 255) / 256, 256, 0, stream>>>(bufA, src, dst, dinv, bufB, N_EDGES);
    sl_bias_relu_kernel<<<(feat4 + 255) / 256, 256, 0, stream>>>(bufB, bufA, dinv, b2, N_NODES);

    // decoder: d_out = bufB @ Wd + bd
    gemm_wmma_f32_kernel<<<gOut, blk, 0, stream>>>(bufB, Wd, (float*)d_out, bd, N_NODES, D_OUTF);
}